// GATLayer_7988639171164
// MI455X (gfx1250) — compile-verified
//
#include <hip/hip_runtime.h>
#include <math.h>

#define N_NODES 50000
#define N_EDGES 800000
#define DIM     128
#define LRELU_A 0.2f

typedef __attribute__((ext_vector_type(2))) float v2f;
typedef __attribute__((ext_vector_type(8))) float v8f;

// ---------------------------------------------------------------------------
// Kernel 1: Wh = h @ W via fp32 WMMA 16x16x4.
// One wave per 16-row strip of Wh; it produces all 8 column tiles (128 cols).
// A (16x4 f32): lanes 0-15 hold rows 0-15 K={0,1}; lanes 16-31 K={2,3}.
// B (4x16 f32): lanes 0-15 hold cols 0-15 K={0,1}; lanes 16-31 K={2,3}.
// C/D (16x16 f32): VGPR i -> row i (lanes 0-15) / row i+8 (lanes 16-31).
// ---------------------------------------------------------------------------
__global__ __launch_bounds__(128) void gat_gemm_wmma(const float* __restrict__ h,
                                                     const float* __restrict__ W,
                                                     float* __restrict__ Wh) {
  const int wave = (int)((blockIdx.x * blockDim.x + threadIdx.x) >> 5);
  const int lane = (int)(threadIdx.x & 31);
  if (wave >= N_NODES / 16) return;

  const int row0  = wave * 16;
  const int rsel  = lane & 15;          // A row / B,C col within tile
  const int khalf = (lane >> 4) * 2;    // 0 for lanes 0-15, 2 for lanes 16-31

  v8f acc[8];
  #pragma unroll
  for (int t = 0; t < 8; ++t) { v8f z = {}; acc[t] = z; }

  for (int k0 = 0; k0 < DIM; k0 += 4) {
    // A tile: 16x4 block of h
    const float* ap = h + (size_t)(row0 + rsel) * DIM + (k0 + khalf);
    v2f a; a.x = ap[0]; a.y = ap[1];
    #pragma unroll
    for (int t = 0; t < 8; ++t) {
      // B tile: 4x16 block of W (row-major, so consecutive K = stride DIM)
      const float* bp = W + (size_t)(k0 + khalf) * DIM + t * 16 + rsel;
      v2f b; b.x = bp[0]; b.y = bp[DIM];
      acc[t] = __builtin_amdgcn_wmma_f32_16x16x4_f32(
          /*neg_a=*/false, a, /*neg_b=*/false, b,
          /*c_mod=*/(short)0, acc[t], /*reuse_a=*/false, /*reuse_b=*/false);
    }
  }

  // Store: VGPR i holds (row0 + i + 8*(lane>>4), col = t*16 + (lane&15))
  const int srow = row0 + (lane >> 4) * 8;
  #pragma unroll
  for (int t = 0; t < 8; ++t) {
    #pragma unroll
    for (int i = 0; i < 8; ++i) {
      Wh[(size_t)(srow + i) * DIM + t * 16 + rsel] = acc[t][i];
    }
  }
}

// ---------------------------------------------------------------------------
// Kernel 2: per-node attention logit halves: s[n]=Wh[n]·a[:128], d[n]=Wh[n]·a[128:]
// One wave32 per node, shuffle reduction.
// ---------------------------------------------------------------------------
__global__ __launch_bounds__(256) void gat_sd(const float* __restrict__ Wh,
                                              const float* __restrict__ a,
                                              float* __restrict__ s,
                                              float* __restrict__ d) {
  const int node = (int)((blockIdx.x * blockDim.x + threadIdx.x) >> 5);
  const int lane = (int)(threadIdx.x & 31);
  if (node >= N_NODES) return;
  const float* row = Wh + (size_t)node * DIM;
  float ss = 0.f, dd = 0.f;
  #pragma unroll
  for (int j = lane; j < DIM; j += 32) {
    float v = row[j];
    ss += v * a[j];
    dd += v * a[DIM + j];
  }
  #pragma unroll
  for (int off = 16; off > 0; off >>= 1) {
    ss += __shfl_down(ss, off, 32);
    dd += __shfl_down(dd, off, 32);
  }
  if (lane == 0) { s[node] = ss; d[node] = dd; }
}

// ---------------------------------------------------------------------------
// Kernel 3: init h_out = h (residual), emax = -inf, denom = 0.
// ---------------------------------------------------------------------------
__global__ void gat_init(const float* __restrict__ h, float* __restrict__ hout,
                         float* __restrict__ emax, float* __restrict__ denom) {
  const size_t i = (size_t)blockIdx.x * blockDim.x + threadIdx.x;
  if (i < (size_t)N_NODES * DIM) hout[i] = h[i];
  if (i < N_NODES) { emax[i] = -INFINITY; denom[i] = 0.f; }
}

// ---------------------------------------------------------------------------
// Kernel 4: edge logits + segment max (two-sided int-trick float atomic max).
// ---------------------------------------------------------------------------
__global__ void gat_edge_logits(const float* __restrict__ s, const float* __restrict__ d,
                                const int* __restrict__ src, const int* __restrict__ dst,
                                float* __restrict__ e, float* __restrict__ emax) {
  const int i = (int)(blockIdx.x * blockDim.x + threadIdx.x);
  if (i >= N_EDGES) return;
  float v = s[src[i]] + d[dst[i]];
  v = (v > 0.f) ? v : LRELU_A * v;          // leaky_relu
  e[i] = v;
  float* addr = emax + dst[i];
  if (v >= 0.f) atomicMax((int*)addr, __float_as_int(v));
  else          atomicMin((unsigned int*)addr, __float_as_uint(v));
}

// ---------------------------------------------------------------------------
// Kernel 5: emax fixup (nodes with no in-edges: -inf -> 0, matching reference).
// ---------------------------------------------------------------------------
__global__ void gat_fix_emax(float* __restrict__ emax) {
  const int i = (int)(blockIdx.x * blockDim.x + threadIdx.x);
  if (i < N_NODES) {
    float m = emax[i];
    if (!isfinite(m)) emax[i] = 0.f;
  }
}

// ---------------------------------------------------------------------------
// Kernel 6: ex = exp(e - emax[dst]); denom[dst] += ex (atomic).
// ---------------------------------------------------------------------------
__global__ void gat_edge_exp(const float* __restrict__ e, const int* __restrict__ dst,
                             const float* __restrict__ emax, float* __restrict__ ex,
                             float* __restrict__ denom) {
  const int i = (int)(blockIdx.x * blockDim.x + threadIdx.x);
  if (i >= N_EDGES) return;
  const float v = expf(e[i] - emax[dst[i]]);
  ex[i] = v;
  atomicAdd(&denom[dst[i]], v);
}

// ---------------------------------------------------------------------------
// Kernel 7: h_out[dst] += (ex/denom[dst]) * Wh[src].
// One wave32 per edge; float4 gather, 4 atomic f32 adds per lane.
// ---------------------------------------------------------------------------
__global__ __launch_bounds__(256) void gat_scatter(const float* __restrict__ Wh,
                                                   const int* __restrict__ src,
                                                   const int* __restrict__ dst,
                                                   const float* __restrict__ ex,
                                                   const float* __restrict__ denom,
                                                   float* __restrict__ hout) {
  const int edge = (int)((blockIdx.x * blockDim.x + threadIdx.x) >> 5);
  const int lane = (int)(threadIdx.x & 31);
  if (edge >= N_EDGES) return;
  const int sn = src[edge];
  const int dn = dst[edge];
  const float w = ex[edge] / denom[dn];
  const float4 v = ((const float4*)(Wh + (size_t)sn * DIM))[lane];
  float* orow = hout + (size_t)dn * DIM + lane * 4;
  atomicAdd(orow + 0, w * v.x);
  atomicAdd(orow + 1, w * v.y);
  atomicAdd(orow + 2, w * v.z);
  atomicAdd(orow + 3, w * v.w);
}

// ---------------------------------------------------------------------------
extern "C" void kernel_launch(void* const* d_in, const int* in_sizes, int n_in,
                              void* d_out, int out_size, void* d_ws, size_t ws_size,
                              hipStream_t stream) {
  const float* h   = (const float*)d_in[0];   // N*DIN
  const float* W   = (const float*)d_in[1];   // DIN*DOUT
  const float* a   = (const float*)d_in[2];   // 2*DOUT
  const int*   src = (const int*)d_in[3];     // E
  const int*   dst = (const int*)d_in[4];     // E
  float* hout = (float*)d_out;                // N*DOUT

  // workspace layout (floats)
  float* ws    = (float*)d_ws;
  float* Wh    = ws;                              // 6,400,000
  float* s     = Wh   + (size_t)N_NODES * DIM;    //    50,000
  float* d     = s    + N_NODES;                  //    50,000
  float* emax  = d    + N_NODES;                  //    50,000
  float* denom = emax + N_NODES;                  //    50,000
  float* e     = denom + N_NODES;                 //   800,000
  float* ex    = e    + N_EDGES;                  //   800,000

  // 1) GEMM: 3125 row-tiles, 4 waves/block of 128 threads
  {
    const int waves = N_NODES / 16;               // 3125
    const int blocks = (waves + 3) / 4;           // 782
    gat_gemm_wmma<<<blocks, 128, 0, stream>>>(h, W, Wh);
  }
  // 2) s, d: one wave per node
  gat_sd<<<(N_NODES * 32 + 255) / 256, 256, 0, stream>>>(Wh, a, s, d);
  // 3) init residual + softmax state
  {
    const size_t total = (size_t)N_NODES * DIM;
    gat_init<<<(unsigned)((total + 255) / 256), 256, 0, stream>>>(h, hout, emax, denom);
  }
  // 4) edge logits + segment max
  gat_edge_logits<<<(N_EDGES + 255) / 256, 256, 0, stream>>>(s, d, src, dst, e, emax);
  // 5) fix -inf
  gat_fix_emax<<<(N_NODES + 255) / 256, 256, 0, stream>>>(emax);
  // 6) exp + segment sum
  gat_edge_exp<<<(N_EDGES + 255) / 256, 256, 0, stream>>>(e, dst, emax, ex, denom);
  // 7) weighted scatter-add (wave per edge, 8 edges per block)
  gat_scatter<<<(N_EDGES * 32 + 255) / 256, 256, 0, stream>>>(Wh, src, dst, ex, denom, hout);
}